// DirectedMPNNLayer_46523085750932
// MI455X (gfx1250) — compile-verified
//
#include <hip/hip_runtime.h>

// ---------------------------------------------------------------------------
// Directed MPNN layer on gfx1250 (MI455X), all-f32 via V_WMMA_F32_16X16X4_F32.
//
// out[e] = relu([x[src_e], edge_attr[e], agg[src_e] - msg[rev(e)]] @ W_upd^T + b_upd)
//   msg[f] = relu(edge_hidden[f] @ W_msg^T + b_msg)
//   agg[v] = sum_{f: tgt_f == v} msg[f]
//   rev(e) = e +/- E/2   (directed reverse-pair construction of edge_index)
// ---------------------------------------------------------------------------

typedef float v2f __attribute__((ext_vector_type(2)));
typedef float v8f __attribute__((ext_vector_type(8)));

// D = A(16x4) * B(4x16) + C(16x16), fp32 WMMA.
// 8-arg pattern: (neg_a, A, neg_b, B, c_mod, C, reuse_a, reuse_b)
__device__ __forceinline__ v8f wmma_f32_4(v2f a, v2f b, v8f c) {
  return __builtin_amdgcn_wmma_f32_16x16x4_f32(false, a, false, b, (short)0, c,
                                               false, false);
}

// Fragment conventions (ISA 7.12.2, wave32):
//  A 16x4 (MxK):  lane L<16 row M=L, lane L+16 same row; elem v holds K = 4s + 2*hi + v
//  B 4x16 (KxN):  lane holds col N = lane&15; elem v holds K = 4s + 2*hi + v
//  C/D 16x16:     VGPR j: rows M = j + 8*hi, lane holds col N = lane&15
// Both A and B k-pairs are consecutive -> 8-byte v2f loads.

__global__ void __launch_bounds__(256) zero_kernel(float4* __restrict__ p, int n4) {
  int i = blockIdx.x * blockDim.x + threadIdx.x;
  if (i < n4) p[i] = make_float4(0.f, 0.f, 0.f, 0.f);
}

// ---------------------------------------------------------------------------
// Kernel 2: msg = relu(EH @ Wm^T + bm); atomic scatter-add rows into agg[tgt].
// One wave handles one 16-edge tile. W_msg^T B-fragments live in VGPRs.
// ---------------------------------------------------------------------------
__global__ void __launch_bounds__(256) msg_scatter_kernel(
    const float* __restrict__ eh, const float* __restrict__ Wm,
    const float* __restrict__ bm, const int* __restrict__ tgt,
    float* __restrict__ agg, int ntiles) {
  const int lane = threadIdx.x & 31;
  const int lo = lane & 15;
  const int hi = (lane >> 4) & 1;
  const int wave = (blockIdx.x * blockDim.x + threadIdx.x) >> 5;
  if (wave >= ntiles) return;  // wave-uniform: EXEC stays all-ones for WMMA
  const int base = wave * 16;

  // B fragments of W_msg^T: B[k][n] = Wm[n][k]
  v2f Bm[16][4];
#pragma unroll
  for (int s = 0; s < 16; ++s)
#pragma unroll
    for (int nb = 0; nb < 4; ++nb)
      Bm[s][nb] = *(const v2f*)(Wm + (nb * 16 + lo) * 64 + 4 * s + 2 * hi);

  float bc[4];
#pragma unroll
  for (int nb = 0; nb < 4; ++nb) bc[nb] = bm[nb * 16 + lo];

  // A fragments: 16 rows of edge_hidden
  v2f A[16];
#pragma unroll
  for (int s = 0; s < 16; ++s)
    A[s] = *(const v2f*)(eh + (size_t)(base + lo) * 64 + 4 * s + 2 * hi);

  v8f acc[4] = {};
#pragma unroll
  for (int s = 0; s < 16; ++s)
#pragma unroll
    for (int nb = 0; nb < 4; ++nb)
      acc[nb] = wmma_f32_4(A[s], Bm[s][nb], acc[nb]);

  int tg[8];
#pragma unroll
  for (int j = 0; j < 8; ++j) tg[j] = tgt[base + j + 8 * hi];

#pragma unroll
  for (int nb = 0; nb < 4; ++nb) {
    const int col = nb * 16 + lo;
#pragma unroll
    for (int j = 0; j < 8; ++j) {
      float v = fmaxf(acc[nb][j] + bc[nb], 0.0f);
      __hip_atomic_fetch_add(agg + (size_t)tg[j] * 64 + col, v,
                             __ATOMIC_RELAXED, __HIP_MEMORY_SCOPE_AGENT);
    }
  }
}

// ---------------------------------------------------------------------------
// Kernel 3: recompute msg for reverse edges, edge_messages = agg[src]-msg_rev,
// then out = relu([x[src], edge_attr, em] @ W_upd^T + b_upd).  4 waves/block.
// ---------------------------------------------------------------------------
__global__ void __launch_bounds__(128) out_kernel(
    const float* __restrict__ x, const float* __restrict__ ea,
    const float* __restrict__ eh, const float* __restrict__ Wm,
    const float* __restrict__ bm, const float* __restrict__ Wu,
    const float* __restrict__ bu, const int* __restrict__ srcv,
    const float* __restrict__ agg, float* __restrict__ out,
    int ntiles, int E2) {
  __shared__ float sW[64 * 144];      // W_upd row-major [h][k]   (36864 B)
  __shared__ float sEm[4 * 16 * 64];  // per-wave em staging      (16384 B)

  const int tid = threadIdx.x;
  // Cooperative stage of W_upd into LDS (float4 granularity).
  for (int i = tid; i < (64 * 144) / 4; i += 128)
    ((float4*)sW)[i] = ((const float4*)Wu)[i];
  __syncthreads();

  const int lane = tid & 31;
  const int lo = lane & 15;
  const int hi = (lane >> 4) & 1;
  const int wib = tid >> 5;
  const int tile = blockIdx.x * 4 + wib;
  if (tile >= ntiles) return;  // wave-uniform
  float* myEm = sEm + wib * (16 * 64);

  const int base = tile * 16;
  const int rbase = (base >= E2) ? (base - E2) : (base + E2);  // reverse tile

  // ---- phase 1: msg of reverse-edge tile (16x64 = EHrev @ Wm^T) ----
  v2f Bm[16][4];
#pragma unroll
  for (int s = 0; s < 16; ++s)
#pragma unroll
    for (int nb = 0; nb < 4; ++nb)
      Bm[s][nb] = *(const v2f*)(Wm + (nb * 16 + lo) * 64 + 4 * s + 2 * hi);

  v2f A[16];
#pragma unroll
  for (int s = 0; s < 16; ++s)
    A[s] = *(const v2f*)(eh + (size_t)(rbase + lo) * 64 + 4 * s + 2 * hi);

  v8f macc[4] = {};
#pragma unroll
  for (int s = 0; s < 16; ++s)
#pragma unroll
    for (int nb = 0; nb < 4; ++nb)
      macc[nb] = wmma_f32_4(A[s], Bm[s][nb], macc[nb]);

  int sidx[8];
#pragma unroll
  for (int j = 0; j < 8; ++j) sidx[j] = srcv[base + j + 8 * hi];

  // em = agg[src] - relu(macc + bm), staged to LDS (C-layout -> row-major).
#pragma unroll
  for (int nb = 0; nb < 4; ++nb) {
    const int col = nb * 16 + lo;
    const float bcv = bm[col];
#pragma unroll
    for (int j = 0; j < 8; ++j) {
      float m = fmaxf(macc[nb][j] + bcv, 0.0f);
      float em = agg[(size_t)sidx[j] * 64 + col] - m;
      myEm[(j + 8 * hi) * 64 + col] = em;
    }
  }
  // Cross-lane LDS RAW within one wave: DS pipe is in-order per wave; fence
  // only against compiler reordering + drain DScnt explicitly.
  asm volatile("s_wait_dscnt 0" ::: "memory");

  // ---- phase 2: out tile = combined(16x144) @ W_upd^T ----
  const int arow = srcv[base + lo];  // x-row gathered by src for this A row
  v8f uacc[4] = {};
#pragma unroll
  for (int s = 0; s < 36; ++s) {
    const int kc = 4 * s + 2 * hi;
    v2f a;
    if (s < 16) {            // cols 0..63  : x[src]
      a = *(const v2f*)(x + (size_t)arow * 64 + kc);
    } else if (s < 20) {     // cols 64..79 : edge_attr
      a = *(const v2f*)(ea + (size_t)(base + lo) * 16 + (kc - 64));
    } else {                 // cols 80..143: edge_messages from LDS
      a = *(const v2f*)(myEm + lo * 64 + (kc - 80));
    }
#pragma unroll
    for (int nb = 0; nb < 4; ++nb) {
      v2f b = *(const v2f*)(sW + (nb * 16 + lo) * 144 + kc);  // ds_load_b64
      uacc[nb] = wmma_f32_4(a, b, uacc[nb]);
    }
  }

#pragma unroll
  for (int nb = 0; nb < 4; ++nb) {
    const int col = nb * 16 + lo;
    const float bv = bu[col];
#pragma unroll
    for (int j = 0; j < 8; ++j) {
      float v = fmaxf(uacc[nb][j] + bv, 0.0f);
      out[(size_t)(base + j + 8 * hi) * 64 + col] = v;
    }
  }
}

extern "C" void kernel_launch(void* const* d_in, const int* in_sizes, int n_in,
                              void* d_out, int out_size, void* d_ws,
                              size_t ws_size, hipStream_t stream) {
  (void)n_in; (void)out_size; (void)ws_size;
  const float* x  = (const float*)d_in[0];
  const float* ea = (const float*)d_in[1];
  const float* eh = (const float*)d_in[2];
  const float* Wm = (const float*)d_in[3];
  const float* bm = (const float*)d_in[4];
  const float* Wu = (const float*)d_in[5];
  const float* bu = (const float*)d_in[6];
  const int*   ei = (const int*)d_in[7];

  const int N = in_sizes[0] / 64;   // 40000
  const int E = in_sizes[7] / 2;    // 800000
  const int E2 = E / 2;
  const int* src = ei;
  const int* tgt = ei + E;
  float* agg = (float*)d_ws;        // N x 64 f32 accumulator (10.24 MB)

  const int n4 = (N * 64) / 4;
  zero_kernel<<<(n4 + 255) / 256, 256, 0, stream>>>((float4*)agg, n4);

  const int ntiles = E / 16;        // 50000 (E and E/2 are multiples of 16)
  msg_scatter_kernel<<<(ntiles + 7) / 8, 256, 0, stream>>>(eh, Wm, bm, tgt,
                                                           agg, ntiles);
  out_kernel<<<(ntiles + 3) / 4, 128, 0, stream>>>(
      x, ea, eh, Wm, bm, Wu, bu, src, agg, (float*)d_out, ntiles, E2);
}